// CopyNet_67791763800710
// MI455X (gfx1250) — compile-verified
//
#include <hip/hip_runtime.h>
#include <hip/hip_bf16.h>
#include <math.h>

#define B_ 32
#define S_ 512
#define H_ 1024
#define T_ 32
#define G3 (3 * H_)

typedef __attribute__((ext_vector_type(2))) float v2f;
typedef __attribute__((ext_vector_type(8))) float v8f;

// ---------------------------------------------------------------------------
// u[d] = sum_h V[h] * W[h*H + d]   (V @ W, W is [H,H] row-major)
// ---------------------------------------------------------------------------
__global__ __launch_bounds__(256) void k_vw(const float* __restrict__ V,
                                            const float* __restrict__ W,
                                            float* __restrict__ out) {
    int d = blockIdx.x * 256 + threadIdx.x;  // 0..H-1
    float acc = 0.f;
    for (int h = 0; h < H_; ++h) acc += V[h] * W[(size_t)h * H_ + d];
    out[d] = acc;
}

// ---------------------------------------------------------------------------
// tv[b,s] = inputs[b,s,:] . w2v   (one block per (b,s))
// ---------------------------------------------------------------------------
__global__ __launch_bounds__(256) void k_tv(const float* __restrict__ inputs,
                                            const float* __restrict__ w2v,
                                            float* __restrict__ tv) {
    int bs = blockIdx.x;
    const float* x = inputs + (size_t)bs * H_;
    float acc = 0.f;
    for (int c = threadIdx.x; c < H_; c += 256) acc += x[c] * w2v[c];
    __shared__ float red[256];
    red[threadIdx.x] = acc;
    __syncthreads();
    for (int off = 128; off > 0; off >>= 1) {
        if (threadIdx.x < off) red[threadIdx.x] += red[threadIdx.x + off];
        __syncthreads();
    }
    if (threadIdx.x == 0) tv[bs] = red[0];
}

__global__ void k_zero(float* __restrict__ p, int n) {
    int i = blockIdx.x * blockDim.x + threadIdx.x;
    if (i < n) p[i] = 0.f;
}

// ---------------------------------------------------------------------------
// Gate GEMMs via V_WMMA_F32_16X16X4_F32:
//   blockIdx.y==0: gi = x @ w_ih^T    blockIdx.y==1: gh = h @ w_hh^T
// Block = 128 threads = 4 waves. Wave w: mTile = w&1, two N tiles at
// blockIdx.x*64 + (w>>1)*32. A fragment loaded once per K step, reused for
// both B fragments.
// ---------------------------------------------------------------------------
__global__ __launch_bounds__(128) void k_gemm_gates(const float* __restrict__ x,
                                                    const float* __restrict__ h,
                                                    const float* __restrict__ w_ih,
                                                    const float* __restrict__ w_hh,
                                                    float* __restrict__ gi,
                                                    float* __restrict__ gh) {
    const int which = blockIdx.y;
    const float* A = which ? h : x;        // [B,H]
    const float* W = which ? w_hh : w_ih;  // [3H,H], used transposed
    float* D = which ? gh : gi;            // [B,3H]

    const int lane  = threadIdx.x & 31;
    const int wave  = threadIdx.x >> 5;
    const int mTile = wave & 1;
    const int nBase = blockIdx.x * 64 + (wave >> 1) * 32;
    const int l15   = lane & 15;
    const int hiHalf = lane >> 4;          // 0 for lanes 0-15, 1 for 16-31
    const int kOff  = hiHalf << 1;         // 0 or 2

    const float* Arow = A + (size_t)(mTile * 16 + l15) * H_;
    const float* Wc0  = W + (size_t)(nBase + l15) * H_;
    const float* Wc1  = W + (size_t)(nBase + 16 + l15) * H_;

    v8f c0 = {};
    v8f c1 = {};
    for (int k = 0; k < H_; k += 4) {
        v2f a, b0, b1;
        a.x  = Arow[k + kOff];
        a.y  = Arow[k + kOff + 1];
        b0.x = Wc0[k + kOff];
        b0.y = Wc0[k + kOff + 1];
        b1.x = Wc1[k + kOff];
        b1.y = Wc1[k + kOff + 1];
        c0 = __builtin_amdgcn_wmma_f32_16x16x4_f32(false, a, false, b0,
                                                   (short)0, c0, false, false);
        c1 = __builtin_amdgcn_wmma_f32_16x16x4_f32(false, a, false, b1,
                                                   (short)0, c1, false, false);
    }
    // C/D layout: lanes 0-15 -> M = v, lanes 16-31 -> M = 8+v; N = lane&15.
    const int mBase = mTile * 16 + hiHalf * 8;
#pragma unroll
    for (int v = 0; v < 8; ++v) {
        int m = mBase + v;
        D[(size_t)m * G3 + nBase + l15]      = c0[v];
        D[(size_t)m * G3 + nBase + 16 + l15] = c1[v];
    }
}

// ---------------------------------------------------------------------------
// GRU elementwise combine (adds biases), in-place hidden update.
// ---------------------------------------------------------------------------
__global__ __launch_bounds__(256) void k_gru(const float* __restrict__ gi,
                                             const float* __restrict__ gh,
                                             const float* __restrict__ b_ih,
                                             const float* __restrict__ b_hh,
                                             float* __restrict__ hidden) {
    int i = blockIdx.x * 256 + threadIdx.x;  // 0..B*H-1
    int b = i >> 10;
    int c = i & (H_ - 1);
    const float* gib = gi + (size_t)b * G3;
    const float* ghb = gh + (size_t)b * G3;
    float ir = gib[c]          + b_ih[c];
    float iz = gib[c + H_]     + b_ih[c + H_];
    float in_ = gib[c + 2*H_]  + b_ih[c + 2*H_];
    float hr = ghb[c]          + b_hh[c];
    float hz = ghb[c + H_]     + b_hh[c + H_];
    float hn = ghb[c + 2*H_]   + b_hh[c + 2*H_];
    float r = 1.f / (1.f + __expf(-(ir + hr)));
    float z = 1.f / (1.f + __expf(-(iz + hz)));
    float n = tanhf(in_ + r * hn);
    hidden[i] = (1.f - z) * n + z * hidden[i];
}

// ---------------------------------------------------------------------------
// Per-batch: td = hidden.u ; scores = tanh(tv+td) -> probs[:,t,:] ;
// argmax(score+mask) ; gather next decoder input.
// ---------------------------------------------------------------------------
__global__ __launch_bounds__(256) void k_score(const float* __restrict__ hidden,
                                               const float* __restrict__ u,
                                               const float* __restrict__ tv,
                                               const float* __restrict__ mask,
                                               const float* __restrict__ inputs,
                                               float* __restrict__ probs,
                                               float* __restrict__ dec_next,
                                               int t) {
    int b = blockIdx.x;
    __shared__ float red[256];
    __shared__ int   redi[256];

    // td = hidden[b,:] . u
    float acc = 0.f;
    const float* hb = hidden + (size_t)b * H_;
    for (int c = threadIdx.x; c < H_; c += 256) acc += hb[c] * u[c];
    red[threadIdx.x] = acc;
    __syncthreads();
    for (int off = 128; off > 0; off >>= 1) {
        if (threadIdx.x < off) red[threadIdx.x] += red[threadIdx.x + off];
        __syncthreads();
    }
    float td = red[0];
    __syncthreads();

    // scores + running argmax of (score + mask)
    float bestv = -INFINITY;
    int   besti = 0;
    float* pout = probs + ((size_t)b * T_ + t) * S_;
    for (int s = threadIdx.x; s < S_; s += 256) {
        float sc = tanhf(tv[b * S_ + s] + td);
        pout[s] = sc;
        float m = sc + mask[b * S_ + s];
        if (m > bestv) { bestv = m; besti = s; }
    }
    red[threadIdx.x]  = bestv;
    redi[threadIdx.x] = besti;
    __syncthreads();
    for (int off = 128; off > 0; off >>= 1) {
        if (threadIdx.x < off) {
            float ov = red[threadIdx.x + off];
            int   oi = redi[threadIdx.x + off];
            if (ov > red[threadIdx.x] ||
                (ov == red[threadIdx.x] && oi < redi[threadIdx.x])) {
                red[threadIdx.x]  = ov;
                redi[threadIdx.x] = oi;
            }
        }
        __syncthreads();
    }
    int idx = redi[0];

    const float* src = inputs + ((size_t)b * S_ + idx) * H_;
    for (int c = threadIdx.x; c < H_; c += 256) dec_next[b * H_ + c] = src[c];
}

// ---------------------------------------------------------------------------
// context[b,:] = softmax(probs[b,0,:] + mask[b,:]) @ inputs[b,:,:]
// ---------------------------------------------------------------------------
__global__ __launch_bounds__(256) void k_context(const float* __restrict__ probs,
                                                 const float* __restrict__ mask,
                                                 const float* __restrict__ inputs,
                                                 float* __restrict__ context) {
    int b = blockIdx.x;
    __shared__ float p[S_];
    __shared__ float red[256];
    const float* sc = probs + (size_t)b * T_ * S_;  // t = 0 row
    float mx = -INFINITY;
    for (int s = threadIdx.x; s < S_; s += 256) {
        float v = sc[s] + mask[b * S_ + s];
        p[s] = v;
        if (v > mx) mx = v;
    }
    red[threadIdx.x] = mx;
    __syncthreads();
    for (int off = 128; off > 0; off >>= 1) {
        if (threadIdx.x < off)
            red[threadIdx.x] = fmaxf(red[threadIdx.x], red[threadIdx.x + off]);
        __syncthreads();
    }
    mx = red[0];
    __syncthreads();
    float sum = 0.f;
    for (int s = threadIdx.x; s < S_; s += 256) {
        float e = __expf(p[s] - mx);
        p[s] = e;
        sum += e;
    }
    red[threadIdx.x] = sum;
    __syncthreads();
    for (int off = 128; off > 0; off >>= 1) {
        if (threadIdx.x < off) red[threadIdx.x] += red[threadIdx.x + off];
        __syncthreads();
    }
    float inv = 1.f / red[0];
    __syncthreads();

    float acc[4] = {0.f, 0.f, 0.f, 0.f};
    for (int s = 0; s < S_; ++s) {
        float w = p[s] * inv;
        const float* xr = inputs + ((size_t)b * S_ + s) * H_;
#pragma unroll
        for (int j = 0; j < 4; ++j) acc[j] += w * xr[threadIdx.x + 256 * j];
    }
#pragma unroll
    for (int j = 0; j < 4; ++j)
        context[b * H_ + threadIdx.x + 256 * j] = acc[j];
}

// ---------------------------------------------------------------------------
extern "C" void kernel_launch(void* const* d_in, const int* in_sizes, int n_in,
                              void* d_out, int out_size, void* d_ws, size_t ws_size,
                              hipStream_t stream) {
    const float* inputs   = (const float*)d_in[0];   // [B,S,H]
    const float* mask     = (const float*)d_in[1];   // [B,S]
    // d_in[2] = inputs_embeds (unused, use_emb=False)
    const float* features = (const float*)d_in[3];   // [B,H]
    const float* w_ih     = (const float*)d_in[4];   // [3H,H]
    const float* b_ih     = (const float*)d_in[5];   // [3H]
    const float* w_hh     = (const float*)d_in[6];   // [3H,H]
    const float* b_hh     = (const float*)d_in[7];   // [3H]
    const float* W1       = (const float*)d_in[8];   // [H,H]
    const float* W2       = (const float*)d_in[9];   // [H,H]
    const float* V        = (const float*)d_in[10];  // [1,H]

    float* out     = (float*)d_out;
    float* probs   = out;                 // [B,T,S]
    float* context = out + (size_t)B_ * T_ * S_;  // [B,H]

    float* ws     = (float*)d_ws;
    float* gi     = ws;                  // B*3H
    float* gh     = gi + B_ * G3;        // B*3H
    float* hidden = gh + B_ * G3;        // B*H
    float* dec    = hidden + B_ * H_;    // B*H
    float* u      = dec + B_ * H_;       // H   (V @ W1)
    float* w2v    = u + H_;              // H   (V @ W2)
    float* tv     = w2v + H_;            // B*S

    // Loop-invariant precompute (the big GEMM collapses to these).
    k_vw<<<dim3(H_ / 256), 256, 0, stream>>>(V, W1, u);
    k_vw<<<dim3(H_ / 256), 256, 0, stream>>>(V, W2, w2v);
    k_tv<<<dim3(B_ * S_), 256, 0, stream>>>(inputs, w2v, tv);
    k_zero<<<dim3((B_ * H_) / 256), 256, 0, stream>>>(hidden, B_ * H_);

    // Sequential GRU/attention recurrence.
    for (int t = 0; t < T_; ++t) {
        const float* xin = (t == 0) ? features : dec;
        k_gemm_gates<<<dim3(G3 / 64, 2), 128, 0, stream>>>(xin, hidden, w_ih,
                                                           w_hh, gi, gh);
        k_gru<<<dim3((B_ * H_) / 256), 256, 0, stream>>>(gi, gh, b_ih, b_hh,
                                                         hidden);
        k_score<<<dim3(B_), 256, 0, stream>>>(hidden, u, tv, mask, inputs,
                                              probs, dec, t);
    }

    // context from step-0 scores.
    k_context<<<dim3(B_), 256, 0, stream>>>(probs, mask, inputs, context);
}